// CoPE_79723182948817
// MI455X (gfx1250) — compile-verified
//
#include <hip/hip_runtime.h>
#include <hip/hip_bf16.h>

typedef float v2f __attribute__((ext_vector_type(2)));
typedef float v8f __attribute__((ext_vector_type(8)));
typedef unsigned int u32x4 __attribute__((ext_vector_type(4)));
typedef int i32x8 __attribute__((ext_vector_type(8)));
typedef int i32x4 __attribute__((ext_vector_type(4)));

#define DDIM   128
#define NPOS   512
#define KLEN   2048
#define TM     16        // q-rows per block
#define WAVES  8
#define BLOCK  256

#define ROWPAD_F (KLEN + KLEN / 64)            // 2080 floats: TDM pads +1 DWORD per 64
#define TABLE_F  (TM * NPOS)                   // 8192 floats = 32 KB
#define LDS_F    (TABLE_F + WAVES * ROWPAD_F)  // 24832 floats = 97 KB (<=320KB/WGP)

// LDS index map matching TDM pad_interval=64 DWORDs, pad_amount=1 DWORD
__device__ __forceinline__ int padix(int g) { return g + (g >> 6); }

__device__ __forceinline__ int rfl(unsigned v) {
  return __builtin_amdgcn_readfirstlane((int)v);
}

// Issue a TDM DMA: one padded row of KLEN floats, global -> LDS.
// D# per CDNA5 ISA 8.3/8.4. NOTE: assumes the single __shared__ array sits at
// LDS offset 0 (standard allocation for a lone shared variable).
__device__ __forceinline__ void tdm_load_row(const float* gsrc, unsigned lds_byte_off) {
  unsigned long long ga = (unsigned long long)(uintptr_t)gsrc;
  u32x4 g0;
  g0[0] = (unsigned)rfl(1u);                                   // count=1, user mode
  g0[1] = (unsigned)rfl(lds_byte_off);                         // lds_addr (bytes)
  g0[2] = (unsigned)rfl((unsigned)(ga & 0xFFFFFFFFull));       // global_addr[31:0]
  g0[3] = (unsigned)rfl((unsigned)((ga >> 32) & 0x1FFFFFFull)  // global_addr[56:32]
                        | (2u << 30));                         // type=2 (image)
  i32x8 g1;
  g1[0] = rfl((2u << 16)      // data_size = 4 bytes
            | (1u << 20)      // pad_enable
            | (5u << 22)      // pad_interval: code 5 = 64 DWORDs
            | (0u << 25));    // pad_amount: code 0 = 1 DWORD
  g1[1] = rfl((unsigned)KLEN << 16);   // tensor_dim0[15:0]=2048 in bits[63:48]
  g1[2] = rfl(1u << 16);               // tensor_dim0[31:16]=0, tensor_dim1[15:0]=1
  g1[3] = rfl((unsigned)KLEN << 16);   // tensor_dim1[31:16]=0, tile_dim0=2048
  g1[4] = rfl(1u);                     // tile_dim1=1, tile_dim2=0
  g1[5] = rfl((unsigned)KLEN);         // tensor_dim0_stride[31:0]=2048
  g1[6] = 0;                           // stride hi / tensor_dim1_stride lo
  g1[7] = 0;
  i32x4 z4 = {0, 0, 0, 0};
  i32x8 z8 = {0, 0, 0, 0, 0, 0, 0, 0};
  // 6-arg form (clang-23 / therock-10.0 headers)
  __builtin_amdgcn_tensor_load_to_lds(g0, g1, z4, z4, z8, 0);
}

__global__ __launch_bounds__(BLOCK)
void cope_fused_kernel(const float* __restrict__ query,    // [BHQ, 128]
                       const float* __restrict__ attn,     // [BHQ, 2048]
                       const float* __restrict__ pos_emb,  // [128, 512]
                       const int*   __restrict__ npos_ptr, // scalar
                       float* __restrict__ out)            // [BHQ, 2048]
{
  __shared__ float lds[LDS_F];
  float* table = lds;                       // [16][512] interpolation tables
  float* rev   = lds + TABLE_F + (threadIdx.x >> 5) * ROWPAD_F;  // wave row buf

  const int tid    = threadIdx.x;
  const int lane   = tid & 31;
  const int wave   = tid >> 5;
  const int laneM  = lane & 15;
  const int laneHi = lane >> 4;
  const int r0     = blockIdx.x * TM;
  const unsigned ldsRowOff = (unsigned)((TABLE_F + wave * ROWPAD_F) * 4);

  const float clampv = (float)(*npos_ptr - 1);   // 511.0f

  // ---- kick off row-0 DMA per wave: hidden behind the whole GEMM phase ----
  tdm_load_row(attn + (size_t)(r0 + wave) * KLEN, ldsRowOff);

  // ---- phase 1: table[16][512] = query(16x128) @ pos_emb(128x512), f32 WMMA ----
  {
    const int n0 = wave * 64;            // each wave owns 64 columns (4 N-tiles)
    const int nA = n0 + 0 * 16 + laneM;
    const int nB = n0 + 1 * 16 + laneM;
    const int nC = n0 + 2 * 16 + laneM;
    const int nD = n0 + 3 * 16 + laneM;
    v8f acc0 = {}, acc1 = {}, acc2 = {}, acc3 = {};
    // A row for this lane: query row (r0+laneM), viewed as float2 pairs
    const float2* aq = (const float2*)(query + (size_t)(r0 + laneM) * DDIM);

    #pragma unroll 4
    for (int kt = 0; kt < 32; ++kt) {
      const int kk = kt * 4 + 2 * laneHi;            // lane K base
      float2 a2 = aq[kt * 2 + laneHi];               // A[M=laneM][kk], [kk+1]
      v2f a; a.x = a2.x; a.y = a2.y;
      const float* b0 = pos_emb + (size_t)kk * NPOS;  // B row K=kk
      const float* b1 = b0 + NPOS;                    // B row K=kk+1
      v2f b;
      b.x = b0[nA]; b.y = b1[nA];
      acc0 = __builtin_amdgcn_wmma_f32_16x16x4_f32(false, a, false, b, (short)0, acc0, false, false);
      b.x = b0[nB]; b.y = b1[nB];
      acc1 = __builtin_amdgcn_wmma_f32_16x16x4_f32(false, a, false, b, (short)0, acc1, false, false);
      b.x = b0[nC]; b.y = b1[nC];
      acc2 = __builtin_amdgcn_wmma_f32_16x16x4_f32(false, a, false, b, (short)0, acc2, false, false);
      b.x = b0[nD]; b.y = b1[nD];
      acc3 = __builtin_amdgcn_wmma_f32_16x16x4_f32(false, a, false, b, (short)0, acc3, false, false);
    }

    // C/D layout: VGPR v -> row (v + 8*laneHi), col = lane%16 within each N tile
    #pragma unroll
    for (int v = 0; v < 8; ++v) {
      float* trow = table + (v + 8 * laneHi) * NPOS;
      trow[nA] = acc0[v];
      trow[nB] = acc1[v];
      trow[nC] = acc2[v];
      trow[nD] = acc3[v];
    }
  }
  __syncthreads();   // tables complete (cross-wave columns)

  // ---- phase 2: sigmoid -> reversed cumsum -> clamp -> gather/interp ----
  for (int it = 0; it < 2; ++it) {
    const int rowLocal = it * 8 + wave;          // 0..15
    const size_t row   = (size_t)(r0 + rowLocal);
    float*       orow  = out + row * KLEN;
    const float* tab   = table + rowLocal * NPOS;

    if (it == 1) {
      // row-0 stage-out LDS reads must finish before DMA overwrites the buffer
      asm volatile("s_wait_dscnt 0x0" ::: "memory");
      tdm_load_row(attn + row * KLEN, ldsRowOff);
    }
    __builtin_amdgcn_s_wait_tensorcnt(0);        // DMA'd row ready in LDS
    asm volatile("" ::: "memory");

    // pass 1: sigmoid (native exp2/rcp), store back, per-lane chunk sum.
    // lane L owns reversed indices [64L,64L+64) -> key pos g descending;
    // bank = (63-t + 31-L) & 63: conflict-free thanks to TDM padding.
    const int gtop = 2047 - lane * 64;
    float s = 0.0f;
    #pragma unroll 8
    for (int t = 0; t < 64; ++t) {
      const int ix = padix(gtop - t);
      float x = rev[ix];
      float e = __builtin_amdgcn_exp2f(-x * 1.44269504088896340736f);
      float g = __builtin_amdgcn_rcpf(1.0f + e);
      rev[ix] = g;
      s += g;
    }

    // wave-level inclusive scan of 32 chunk sums -> exclusive prefix
    float mySum = s;
    #pragma unroll
    for (int off = 1; off < 32; off <<= 1) {
      float n = __shfl_up(s, off, 32);
      if (lane >= off) s += n;
    }
    float run = s - mySum;

    // pass 2: running suffix-sum position, clamp, LDS table gather + lerp;
    // result overwrites the buffer at its key position
    #pragma unroll 4
    for (int t = 0; t < 64; ++t) {
      const int ix = padix(gtop - t);
      float g = rev[ix];
      run += g;
      float pos = fminf(run, clampv);
      float pf  = floorf(pos);
      float w   = pos - pf;
      int   fi  = (int)pf;
      int   ci  = (int)ceilf(pos);
      float val = tab[ci] * w + tab[fi] * (1.0f - w);
      rev[ix] = val;
    }

    // stage out: padded-LDS gather (pad const within aligned quad), coalesced
    // float4 global stores
    #pragma unroll 4
    for (int j = 0; j < 16; ++j) {
      const int g0 = j * 128 + lane * 4;
      const int p0 = padix(g0);
      float4 o;
      o.x = rev[p0 + 0];
      o.y = rev[p0 + 1];
      o.z = rev[p0 + 2];
      o.w = rev[p0 + 3];
      *(float4*)(orow + g0) = o;
    }
  }
}

extern "C" void kernel_launch(void* const* d_in, const int* in_sizes, int n_in,
                              void* d_out, int out_size, void* d_ws, size_t ws_size,
                              hipStream_t stream) {
  const float* query   = (const float*)d_in[0];   // [2,12,2048,128] f32
  const float* attn    = (const float*)d_in[1];   // [2,12,2048,2048] f32
  const float* pos_emb = (const float*)d_in[2];   // [1,128,512] f32
  const int*   npos    = (const int*)d_in[3];     // scalar int
  float* out = (float*)d_out;                      // [2,12,2048,2048] f32

  const int rows = in_sizes[1] / KLEN;             // B*H*Q = 49152
  const int grid = rows / TM;                      // 3072 blocks
  cope_fused_kernel<<<grid, BLOCK, 0, stream>>>(query, attn, pos_emb, npos, out);
}